// DomainAdjustedChebyshevConv_42502996361884
// MI455X (gfx1250) — compile-verified
//
#include <hip/hip_runtime.h>

typedef float v2f __attribute__((ext_vector_type(2)));
typedef float v8f __attribute__((ext_vector_type(8)));

#define DFEAT 128

// ---------------- elementwise / setup kernels ----------------

__global__ void zero_f4_kernel(float* __restrict__ p, int n4) {
    int i = blockIdx.x * blockDim.x + threadIdx.x;
    if (i < n4) ((float4*)p)[i] = make_float4(0.f, 0.f, 0.f, 0.f);
}

__global__ void deg_kernel(const int* __restrict__ dst, float* __restrict__ deg, int E) {
    int e = blockIdx.x * blockDim.x + threadIdx.x;
    if (e < E) atomicAdd(&deg[dst[e]], 1.0f);
}

__global__ void dinv_kernel(const float* __restrict__ deg, float* __restrict__ dinv, int N) {
    int i = blockIdx.x * blockDim.x + threadIdx.x;
    if (i < N) dinv[i] = 1.0f / sqrtf(fmaxf(deg[i], 1.0f));
}

__global__ void h_init_kernel(const float* __restrict__ feat, const float* __restrict__ theta,
                              float* __restrict__ h, int n4) {
    int i = blockIdx.x * blockDim.x + threadIdx.x;
    if (i >= n4) return;
    float t0 = theta[0];
    float4 f = ((const float4*)feat)[i];
    ((float4*)h)[i] = make_float4(t0 * f.x, t0 * f.y, t0 * f.z, t0 * f.w);
}

// ---------------- edge scatter: agg[dst] += dinv[src] * x[src] ----------------
// One wave (32 lanes) per edge; each lane moves 4 consecutive floats (float4
// gather + 4 non-returning global_atomic_add_f32). Working set is L2-resident.

__global__ void scatter_kernel(const float* __restrict__ x, const float* __restrict__ dinv,
                               const int* __restrict__ src, const int* __restrict__ dst,
                               float* __restrict__ agg, int E) {
    int t = blockIdx.x * blockDim.x + threadIdx.x;
    int e = t >> 5;
    int lane = t & 31;
    if (e >= E) return;
    int s = src[e];
    int d = dst[e];
    float w = dinv[s];
    float4 v = *(const float4*)(x + (size_t)s * DFEAT + lane * 4);
    float* ap = agg + (size_t)d * DFEAT + lane * 4;
    atomicAdd(ap + 0, w * v.x);
    atomicAdd(ap + 1, w * v.y);
    atomicAdd(ap + 2, w * v.z);
    atomicAdd(ap + 3, w * v.w);
}

// ---------------- fused Chebyshev update ----------------
// T_k = c2 * L'(x) - T_{k-2}  with  L'(x) = -dinv * agg   =>  t = -c2*dinv*agg - tprev
// h  += theta[order] * (T_k + 1) * 0.5 ; also re-zeros agg for the next propagation.

__global__ void cheb_update_kernel(float* __restrict__ agg, const float* __restrict__ dinv,
                                   const float* __restrict__ Tprev, float* __restrict__ Tnew,
                                   float* __restrict__ h, const float* __restrict__ theta,
                                   int order, float c2, int N) {
    int i = blockIdx.x * blockDim.x + threadIdx.x;  // one thread per 4 floats
    int total = N * (DFEAT / 4);
    if (i >= total) return;
    int n = i / (DFEAT / 4);
    float s = -c2 * dinv[n];
    float th = theta[order] * 0.5f;

    float4 a = ((const float4*)agg)[i];
    ((float4*)agg)[i] = make_float4(0.f, 0.f, 0.f, 0.f);  // ready for next scatter

    float4 tp = Tprev ? ((const float4*)Tprev)[i] : make_float4(0.f, 0.f, 0.f, 0.f);
    float4 t;
    t.x = fmaf(s, a.x, -tp.x);
    t.y = fmaf(s, a.y, -tp.y);
    t.z = fmaf(s, a.z, -tp.z);
    t.w = fmaf(s, a.w, -tp.w);
    ((float4*)Tnew)[i] = t;

    float4 hv = ((const float4*)h)[i];
    hv.x = fmaf(th, t.x + 1.0f, hv.x);
    hv.y = fmaf(th, t.y + 1.0f, hv.y);
    hv.z = fmaf(th, t.z + 1.0f, hv.z);
    hv.w = fmaf(th, t.w + 1.0f, hv.w);
    ((float4*)h)[i] = hv;
}

// ---------------- out = leaky_relu(h @ W^T) via V_WMMA_F32_16X16X4_F32 ----------------
// One wave per 16x16 output tile, K=128 in 32 steps of 4. Full fp32 precision.
// A-frag (16x4): lane -> A[m = lane&15][kb], A[m][kb+1], kb = 2*(lane>>4)  (float2 from h row)
// B-frag (4x16): B[k][n] = W[n][k] -> same float2 pattern on W row (n = lane&15)
// C/D: VGPR v -> row v + 8*(lane>>4), col lane&15.

__global__ void gemm_wmma_kernel(const float* __restrict__ h, const float* __restrict__ W,
                                 float* __restrict__ out, int N) {
    const int lane = threadIdx.x & 31;
    const int wave = threadIdx.x >> 5;
    const int tile = blockIdx.x * (blockDim.x >> 5) + wave;
    const int nRowTiles = N >> 4;                // N = 50000 = 3125*16 exactly
    const int row = tile >> 3;                   // 128/16 = 8 column tiles
    const int col = tile & 7;
    if (row >= nRowTiles) return;                // wave-uniform: EXEC stays all-ones

    const int rowBase = row << 4;
    const int colBase = col << 4;
    const int m = lane & 15;
    const int half = lane >> 4;
    const int kb = half << 1;

    const float* hrow = h + (size_t)(rowBase + m) * DFEAT + kb;
    const float* wrow = W + (size_t)(colBase + m) * DFEAT + kb;

    v8f acc = {};
#pragma unroll
    for (int kk = 0; kk < DFEAT; kk += 4) {
        v2f a = *(const v2f*)(hrow + kk);
        v2f b = *(const v2f*)(wrow + kk);
        acc = __builtin_amdgcn_wmma_f32_16x16x4_f32(
            /*neg_a=*/false, a, /*neg_b=*/false, b,
            /*c_mod=*/(short)0, acc, /*reuse_a=*/false, /*reuse_b=*/false);
    }

#pragma unroll
    for (int v = 0; v < 8; ++v) {
        int r = rowBase + v + (half << 3);
        float val = acc[v];
        val = (val >= 0.0f) ? val : 0.01f * val;
        out[(size_t)r * DFEAT + colBase + m] = val;
    }
}

// ---------------- launch ----------------

extern "C" void kernel_launch(void* const* d_in, const int* in_sizes, int n_in,
                              void* d_out, int out_size, void* d_ws, size_t ws_size,
                              hipStream_t stream) {
    const float* feat  = (const float*)d_in[0];   // N x 128
    const int*   src   = (const int*)d_in[1];     // E
    const int*   dst   = (const int*)d_in[2];     // E
    const float* theta = (const float*)d_in[3];   // K+1 = 4
    const float* W     = (const float*)d_in[4];   // 128 x 128
    float* out = (float*)d_out;                   // N x 128

    const int N = in_sizes[0] / DFEAT;            // 50000
    const int E = in_sizes[1];                    // 800000

    float* ws   = (float*)d_ws;
    float* deg  = ws;                              // N
    float* dinv = deg + N;                         // N
    float* agg  = dinv + N;                        // N*128
    float* Ta   = agg + (size_t)N * DFEAT;         // N*128
    float* Tb   = Ta + (size_t)N * DFEAT;          // N*128
    float* h    = Tb + (size_t)N * DFEAT;          // N*128

    const int TB = 256;
    const int nd4 = N * (DFEAT / 4);
    auto cdiv = [](long long a, long long b) { return (int)((a + b - 1) / b); };

    // degrees + dinv
    zero_f4_kernel<<<cdiv(N / 4, TB), TB, 0, stream>>>(deg, N / 4);
    zero_f4_kernel<<<cdiv(nd4, TB), TB, 0, stream>>>(agg, nd4);
    deg_kernel<<<cdiv(E, TB), TB, 0, stream>>>(dst, deg, E);
    dinv_kernel<<<cdiv(N, TB), TB, 0, stream>>>(deg, dinv, N);

    // h = theta0 * feat
    h_init_kernel<<<cdiv(nd4, TB), TB, 0, stream>>>(feat, theta, h, nd4);

    const long long edgeThreads = (long long)E * 32;

    // order 1: T1 = L'(feat)
    scatter_kernel<<<cdiv(edgeThreads, TB), TB, 0, stream>>>(feat, dinv, src, dst, agg, E);
    cheb_update_kernel<<<cdiv(nd4, TB), TB, 0, stream>>>(agg, dinv, nullptr, Ta, h, theta, 1, 1.0f, N);

    // order 2: T2 = 2 L'(T1) - feat
    scatter_kernel<<<cdiv(edgeThreads, TB), TB, 0, stream>>>(Ta, dinv, src, dst, agg, E);
    cheb_update_kernel<<<cdiv(nd4, TB), TB, 0, stream>>>(agg, dinv, feat, Tb, h, theta, 2, 2.0f, N);

    // order 3: T3 = 2 L'(T2) - T1   (overwrite Ta; elementwise in-place safe)
    scatter_kernel<<<cdiv(edgeThreads, TB), TB, 0, stream>>>(Tb, dinv, src, dst, agg, E);
    cheb_update_kernel<<<cdiv(nd4, TB), TB, 0, stream>>>(agg, dinv, Ta, Ta, h, theta, 3, 2.0f, N);

    // out = leaky_relu(h @ W^T) — fp32 WMMA, one wave per 16x16 tile
    const int tiles = (N / 16) * (DFEAT / 16);    // 25000
    gemm_wmma_kernel<<<cdiv(tiles, 256 / 32), 256, 0, stream>>>(h, W, out, N);
}